// hash_encoder_36687610642576
// MI455X (gfx1250) — compile-verified
//
#include <hip/hip_runtime.h>
#include <stdint.h>

// ---------------------------------------------------------------------------
// 2D multiresolution hash-grid encoder (instant-NGP style), MI455X / gfx1250.
//
// Strategy:
//  * All 10 embedding tables total 29,346 entries x 8B = 229.3 KB -> they fit
//    entirely in CDNA5's 320 KB LDS. Stage them once per workgroup with
//    CDNA5 async global->LDS copies (ASYNCcnt path), then every corner gather
//    is a ds_load_b64.
//  * 1024-thread blocks (32 wave32s) = one block per WGP at this LDS size.
//  * Dense-level indices derived exactly from the already-computed floor
//    quotients (no extra divides), hashed levels use the reference hash.
// ---------------------------------------------------------------------------

#define NLEV 10
#define TBL_TOTAL 29346   // sum of table entry counts
#define HASH_MASK 4095u
#define HASH_C2 2654435761u

constexpr int kEntries[NLEV] = {441, 729, 1296, 2304, 4096, 4096, 4096, 4096, 4096, 4096};
constexpr int kOffset [NLEV] = {0, 441, 1170, 2466, 4770, 8866, 12962, 17058, 21154, 25250};

struct TablePtrs { const float* t[NLEV]; };

__device__ __forceinline__ int imin(int a, int b) { return a < b ? a : b; }

__device__ __forceinline__ uint32_t hash2(int ix, int iy) {
  // reference: (x*1u) ^ (y*2654435761u), masked to 12 bits
  uint32_t h = (uint32_t)ix ^ ((uint32_t)iy * HASH_C2);
  return h & HASH_MASK;
}

template <int NL, int BW, bool DENSE, int OFF>
__device__ __forceinline__ void level_embed(float x, float y, const float2* lut,
                                            float& o0, float& o1) {
  const float bwf = (float)BW;
  // exact IEEE division (clang folds /1 and /2); floor quotients reused below
  float qx = floorf(x / bwf);
  float qy = floorf(y / bwf);
  float xm = fmaxf(0.0f, qx * bwf);
  float ym = fmaxf(0.0f, qy * bwf);
  float xM = fminf(320.0f, xm + bwf);
  float yM = fminf(320.0f, ym + bwf);

  int idx0, idx1, idx2, idx3;
  if (DENSE) {
    // floor(xm/bw)==qx and floor(xM/bw)==min(qx+1, 320/BW) exactly
    constexpr int SMAX = 320 / BW;
    int sx0 = (int)qx, sy0 = (int)qy;
    int sx1 = imin(sx0 + 1, SMAX);
    int sy1 = imin(sy0 + 1, SMAX);
    idx0 = sy0 * NL + sx0;
    idx1 = sy0 * NL + sx1;
    idx2 = sy1 * NL + sx0;
    idx3 = sy1 * NL + sx1;
  } else {
    int ixm = (int)xm, iym = (int)ym, ixM = (int)xM, iyM = (int)yM;
    idx0 = (int)hash2(ixm, iym);
    idx1 = (int)hash2(ixM, iym);
    idx2 = (int)hash2(ixm, iyM);
    idx3 = (int)hash2(ixM, iyM);
  }

  float dx0 = xm - x, dy0 = ym - y;
  float dx1 = xM - x, dy1 = yM - y;
  float d0 = sqrtf(dx0 * dx0 + dy0 * dy0);
  float d1 = sqrtf(dx1 * dx1 + dy0 * dy0);
  float d2 = sqrtf(dx0 * dx0 + dy1 * dy1);
  float d3 = sqrtf(dx1 * dx1 + dy1 * dy1);
  float S = (d0 + d1) + (d2 + d3);
  float invS = 1.0f / S;

  float2 e0 = lut[OFF + idx0];   // ds_load_b64 gathers from LDS
  float2 e1 = lut[OFF + idx1];
  float2 e2 = lut[OFF + idx2];
  float2 e3 = lut[OFF + idx3];

  float w0 = 1.0f - d0 * invS;
  float w1 = 1.0f - d1 * invS;
  float w2 = 1.0f - d2 * invS;
  float w3 = 1.0f - d3 * invS;

  o0 = w0 * e0.x + w1 * e1.x + w2 * e2.x + w3 * e3.x;
  o1 = w0 * e0.y + w1 * e1.y + w2 * e2.y + w3 * e3.y;
}

__global__ __launch_bounds__(1024)
void hash_encoder_kernel(const float* __restrict__ pts, TablePtrs tp,
                         float* __restrict__ out, int npts) {
  // 229.3 KB of static LDS — legal on CDNA5 (320 KB per workgroup)
  __shared__ float2 lut[TBL_TOTAL];

  // ---- Stage all tables into LDS via CDNA5 async global->LDS DMA ----------
  const uint32_t ldsbase = (uint32_t)(uintptr_t)(&lut[0]);
  #pragma unroll
  for (int t = 0; t < NLEV; ++t) {
    const uint64_t gbase = (uint64_t)(uintptr_t)tp.t[t];
    const uint32_t lbase = ldsbase + (uint32_t)kOffset[t] * 8u;
    for (int k = (int)threadIdx.x; k < kEntries[t]; k += (int)blockDim.x) {
      uint32_t laddr = lbase + (uint32_t)k * 8u;   // per-lane LDS byte address
      uint32_t goff  = (uint32_t)k * 8u;           // per-lane global byte offset
      // GVS mode: mem = SGPR64 base + VGPR32 offset ; LDS dest in VDST vgpr.
      asm volatile("global_load_async_to_lds_b64 %0, %1, %2"
                   :: "v"(laddr), "v"(goff), "s"(gbase)
                   : "memory");
    }
  }
#if __has_builtin(__builtin_amdgcn_s_wait_asynccnt)
  __builtin_amdgcn_s_wait_asynccnt(0);
#else
  asm volatile("s_wait_asynccnt 0" ::: "memory");
#endif
  __syncthreads();  // all waves' async LDS writes complete & visible

  // ---- Main grid-stride loop over points ----------------------------------
  const int stride = (int)(gridDim.x * blockDim.x);
  for (int p = (int)(blockIdx.x * blockDim.x + threadIdx.x); p < npts; p += stride) {
    const float* pp = pts + (size_t)p * 3;
    float x = pp[0];
    float y = pp[1];
    float z = pp[2];

    float o[21];
    level_embed<16,  20, true,      0>(x, y, lut, o[0],  o[1]);
    level_embed<22,  14, true,    441>(x, y, lut, o[2],  o[3]);
    level_embed<31,  10, true,   1170>(x, y, lut, o[4],  o[5]);
    level_embed<43,   7, true,   2466>(x, y, lut, o[6],  o[7]);
    level_embed<60,   5, false,  4770>(x, y, lut, o[8],  o[9]);
    level_embed<84,   3, false,  8866>(x, y, lut, o[10], o[11]);
    level_embed<117,  2, false, 12962>(x, y, lut, o[12], o[13]);
    level_embed<164,  1, false, 17058>(x, y, lut, o[14], o[15]);
    level_embed<229,  1, false, 21154>(x, y, lut, o[16], o[17]);
    {
      // level 9 (n == NMAX): direct hash of truncated point coords
      uint32_t h = hash2((int)x, (int)y);
      float2 e = lut[25250 + (int)h];
      o[18] = e.x;
      o[19] = e.y;
    }
    o[20] = z;

    float* op = out + (size_t)p * 21;
    #pragma unroll
    for (int k = 0; k < 21; ++k) op[k] = o[k];
  }
}

extern "C" void kernel_launch(void* const* d_in, const int* in_sizes, int n_in,
                              void* d_out, int out_size, void* d_ws, size_t ws_size,
                              hipStream_t stream) {
  (void)d_ws; (void)ws_size; (void)out_size;

  const float* pts = (const float*)d_in[0];
  TablePtrs tp;
  if (n_in >= 1 + NLEV) {
    // tables arrive as 10 separate inputs
    for (int i = 0; i < NLEV; ++i) tp.t[i] = (const float*)d_in[1 + i];
  } else {
    // fallback: tables concatenated flat in one buffer
    const float* flat = (const float*)d_in[1];
    for (int i = 0; i < NLEV; ++i) tp.t[i] = flat + (size_t)kOffset[i] * 2;
  }
  int npts = in_sizes[0] / 3;

  dim3 grid(512), block(1024);
  hipLaunchKernelGGL(hash_encoder_kernel, grid, block, 0, stream,
                     pts, tp, (float*)d_out, npts);
}